// SSMBlock_43542378446998
// MI455X (gfx1250) — compile-verified
//
#include <hip/hip_runtime.h>
#include <math.h>

// ---------------------------------------------------------------------------
// SSM block for MI455X (gfx1250, wave32).
//  Phase 1: x@W+b via V_WMMA_F32_16X16X4_F32 (full f32 precision), fused
//           softplus/gating + skip (x·D) computed off the A fragments.
//           Two 16-row M tiles per wave so each set of B fragments (from W)
//           feeds 6 WMMAs -> better WMMA:VMEM issue ratio.
//  Phase 2-4: chunked parallel linear recurrence (64 chunks of 64 steps).
// ---------------------------------------------------------------------------

typedef float v2f __attribute__((ext_vector_type(2)));
typedef float v8f __attribute__((ext_vector_type(8)));

#define BATCH   8
#define LSEQ    4096
#define DM      1024
#define SDIM    16
#define NROWS   (BATCH * LSEQ)      // 32768
#define CHUNK   64
#define NCHUNK  (LSEQ / CHUNK)      // 64
#define NINST   (BATCH * NCHUNK)    // 512 chunk instances

__device__ __forceinline__ float softplusf(float v) {
    // numerically stable log1p(exp(v))
    return fmaxf(v, 0.0f) + log1pf(expf(-fabsf(v)));
}

// ---------------------------------------------------------------------------
// Kernel 1: per wave, TWO 16-row M tiles, 3 N tiles (dt | B | C), K step 4.
//   A frag 16x4 f32: lane m=lane&15 (row), kg=lane>>4 selects K pair {2kg,2kg+1}
//   B frag 4x16 f32: lane n=lane&15 (col), kg selects same K pair
//   C/D 16x16 f32  : VGPR v holds row v + 8*(lane>>4), col = lane&15
// ---------------------------------------------------------------------------
__global__ __launch_bounds__(256)
void ssm_gemm_proj_kernel(const float* __restrict__ x,
                          const float* __restrict__ W,
                          const float* __restrict__ bias,
                          const float* __restrict__ Dvec,
                          float* __restrict__ U,     // dt * B   (NROWS, 16)
                          float* __restrict__ CP,    // C        (NROWS, 16)
                          float* __restrict__ SKIP)  // x . D    (NROWS)
{
    const int lane = threadIdx.x & 31;
    const int wave = threadIdx.x >> 5;
    const int rowBase = (blockIdx.x * 8 + wave) * 32;   // 2 M tiles per wave
    const int m  = lane & 15;
    const int kg = lane >> 4;             // which K pair of the 16x4 tile

    const float* __restrict__ xRow0 = x + (size_t)(rowBase +      m) * DM;
    const float* __restrict__ xRow1 = x + (size_t)(rowBase + 16 + m) * DM;

    v8f c00 = {}, c01 = {}, c02 = {};     // tile 0 accumulators (dt|B|C)
    v8f c10 = {}, c11 = {}, c12 = {};     // tile 1 accumulators
    float sacc0 = 0.0f, sacc1 = 0.0f;     // partials of x . D

    for (int kk = 0; kk < DM; kk += 4) {
        const int k0 = kk + 2 * kg;
        v2f a0 = *(const v2f*)(xRow0 + k0);          // A fragments (b64)
        v2f a1 = *(const v2f*)(xRow1 + k0);
        v2f d  = *(const v2f*)(Dvec + k0);
        sacc0 = fmaf(a0.x, d.x, sacc0);
        sacc0 = fmaf(a0.y, d.y, sacc0);
        sacc1 = fmaf(a1.x, d.x, sacc1);
        sacc1 = fmaf(a1.y, d.y, sacc1);
        __builtin_prefetch(xRow0 + kk + 128, 0, 1);  // global_prefetch_b8
        __builtin_prefetch(xRow1 + kk + 128, 0, 1);

        const float* __restrict__ w0 = W + (size_t)k0 * 48 + m;  // row k0
        const float* __restrict__ w1 = w0 + 48;                  // row k0+1
        v2f b0 = { w0[ 0], w1[ 0] };   // N tile 0 : dt params
        v2f b1 = { w0[16], w1[16] };   // N tile 1 : B params
        v2f b2 = { w0[32], w1[32] };   // N tile 2 : C params

        c00 = __builtin_amdgcn_wmma_f32_16x16x4_f32(false, a0, false, b0,
                                                    (short)0, c00, false, false);
        c01 = __builtin_amdgcn_wmma_f32_16x16x4_f32(false, a0, false, b1,
                                                    (short)0, c01, false, false);
        c02 = __builtin_amdgcn_wmma_f32_16x16x4_f32(false, a0, false, b2,
                                                    (short)0, c02, false, false);
        c10 = __builtin_amdgcn_wmma_f32_16x16x4_f32(false, a1, false, b0,
                                                    (short)0, c10, false, false);
        c11 = __builtin_amdgcn_wmma_f32_16x16x4_f32(false, a1, false, b1,
                                                    (short)0, c11, false, false);
        c12 = __builtin_amdgcn_wmma_f32_16x16x4_f32(false, a1, false, b2,
                                                    (short)0, c12, false, false);
    }

    // lanes L and L+16 hold the two K-halves of the same row's dot with D
    sacc0 += __shfl_xor(sacc0, 16, 32);
    sacc1 += __shfl_xor(sacc1, 16, 32);
    if (kg == 0) {
        SKIP[rowBase +      m] = sacc0;
        SKIP[rowBase + 16 + m] = sacc1;
    }

    const int s  = lane & 15;
    const float bd = bias[s], bb = bias[16 + s], bc = bias[32 + s];
    #pragma unroll
    for (int v = 0; v < 8; ++v) {
        // tile 0
        int row = rowBase + v + 8 * kg;
        float dt = softplusf(c00[v] + bd);
        U [row * SDIM + s] = dt * (c01[v] + bb);
        CP[row * SDIM + s] = c02[v] + bc;
        // tile 1
        row += 16;
        dt = softplusf(c10[v] + bd);
        U [row * SDIM + s] = dt * (c11[v] + bb);
        CP[row * SDIM + s] = c12[v] + bc;
    }
}

// ---------------------------------------------------------------------------
// Kernel 2: local scan of each 64-step chunk starting from h=0; store endpoint.
// One (b,chunk) instance per 16-lane group.
// ---------------------------------------------------------------------------
__global__ __launch_bounds__(256)
void ssm_chunk_summary_kernel(const float* __restrict__ U,
                              const float* __restrict__ A_log,
                              float* __restrict__ V)   // (NINST, 16)
{
    const int t = blockIdx.x * blockDim.x + threadIdx.x;
    const int s = t & 15;
    const int g = t >> 4;                 // chunk instance id
    const int b = g / NCHUNK;
    const int c = g % NCHUNK;
    const float A = -expf(A_log[s]);

    float h = 0.0f;
    size_t base = ((size_t)(b * LSEQ + c * CHUNK)) * SDIM + s;
    #pragma unroll 4
    for (int i = 0; i < CHUNK; ++i)
        h = fmaf(A, h, U[base + (size_t)i * SDIM]);
    V[g * SDIM + s] = h;
}

// ---------------------------------------------------------------------------
// Kernel 3: cross-chunk prefix:  h_start(c+1) = A^CHUNK * h_start(c) + V(c).
// 128 independent (b,s) lanes, 64 sequential (cheap) steps.
// ---------------------------------------------------------------------------
__global__ __launch_bounds__(128)
void ssm_chunk_prefix_kernel(const float* __restrict__ V,
                             const float* __restrict__ A_log,
                             float* __restrict__ H0)   // (NINST, 16)
{
    const int s = threadIdx.x & 15;
    const int b = threadIdx.x >> 4;
    const float A = -expf(A_log[s]);
    float p = A;                          // A^64 by repeated squaring
    #pragma unroll
    for (int j = 0; j < 6; ++j) p *= p;

    float h = 0.0f;
    for (int c = 0; c < NCHUNK; ++c) {
        H0[(b * NCHUNK + c) * SDIM + s] = h;
        h = fmaf(p, h, V[(b * NCHUNK + c) * SDIM + s]);
    }
}

// ---------------------------------------------------------------------------
// Kernel 4: replay each chunk from its true h_start, emit y_t = C·h + skip.
// Cross-s reduction via 4-step shfl_xor butterfly within each 16-lane group.
// ---------------------------------------------------------------------------
__global__ __launch_bounds__(256)
void ssm_scan_output_kernel(const float* __restrict__ U,
                            const float* __restrict__ CP,
                            const float* __restrict__ H0,
                            const float* __restrict__ SKIP,
                            const float* __restrict__ A_log,
                            float* __restrict__ out)
{
    const int t = blockIdx.x * blockDim.x + threadIdx.x;
    const int s = t & 15;
    const int g = t >> 4;
    const int b = g / NCHUNK;
    const int c = g % NCHUNK;
    const float A = -expf(A_log[s]);

    float h = H0[g * SDIM + s];
    const size_t rbase = (size_t)(b * LSEQ + c * CHUNK);
    for (int i = 0; i < CHUNK; ++i) {
        const size_t idx = (rbase + i) * SDIM + s;
        h = fmaf(A, h, U[idx]);
        float y = CP[idx] * h;
        y += __shfl_xor(y, 1, 32);
        y += __shfl_xor(y, 2, 32);
        y += __shfl_xor(y, 4, 32);
        y += __shfl_xor(y, 8, 32);
        if (s == 0) out[rbase + i] = y + SKIP[rbase + i];
    }
}

// ---------------------------------------------------------------------------
extern "C" void kernel_launch(void* const* d_in, const int* in_sizes, int n_in,
                              void* d_out, int out_size, void* d_ws, size_t ws_size,
                              hipStream_t stream) {
    const float* x     = (const float*)d_in[0];   // (8, 4096, 1024)
    const float* A_log = (const float*)d_in[1];   // (16)
    const float* W     = (const float*)d_in[2];   // (1024, 48)
    const float* bias  = (const float*)d_in[3];   // (48)
    const float* Dvec  = (const float*)d_in[4];   // (1024)
    float* out = (float*)d_out;                   // (8, 4096)

    float* U    = (float*)d_ws;                   // NROWS*16
    float* CP   = U    + (size_t)NROWS * SDIM;    // NROWS*16
    float* SKIP = CP   + (size_t)NROWS * SDIM;    // NROWS
    float* V    = SKIP + (size_t)NROWS;           // NINST*16
    float* H0   = V    + (size_t)NINST * SDIM;    // NINST*16

    // 1024 wave-tasks of 32 rows each, 8 waves per 256-thread block
    ssm_gemm_proj_kernel<<<NROWS / (32 * 8), 256, 0, stream>>>(
        x, W, bias, Dvec, U, CP, SKIP);

    ssm_chunk_summary_kernel<<<(NINST * SDIM) / 256, 256, 0, stream>>>(
        U, A_log, V);

    ssm_chunk_prefix_kernel<<<1, 128, 0, stream>>>(V, A_log, H0);

    ssm_scan_output_kernel<<<(NINST * SDIM) / 256, 256, 0, stream>>>(
        U, CP, H0, SKIP, A_log, out);
}